// GC_withres_39195871544107
// MI455X (gfx1250) — compile-verified
//
#include <hip/hip_runtime.h>
#include <hip/hip_bf16.h>

typedef float v2f __attribute__((ext_vector_type(2)));
typedef float v8f __attribute__((ext_vector_type(8)));

#define N_NODES 50000
#define N_EDGES 800000
#define D 64
#define N_TILES (N_NODES / 16)   // 3125, exact
#define SMOOTHF 0.5f

// ---------------- degree pipeline ----------------

__global__ void k_zero_deg(float* __restrict__ deg) {
    int i = blockIdx.x * blockDim.x + threadIdx.x;
    if (i < N_NODES) deg[i] = 0.0f;
}

__global__ void k_degree(const int* __restrict__ ecol, float* __restrict__ deg) {
    int e = blockIdx.x * blockDim.x + threadIdx.x;
    if (e < N_EDGES) atomicAdd(&deg[ecol[e]], 1.0f);
}

__global__ void k_rsqrt(float* __restrict__ deg) {
    int i = blockIdx.x * blockDim.x + threadIdx.x;
    if (i < N_NODES) deg[i] = rsqrtf(deg[i] + 1.0f);   // deg becomes d = (deg+1)^-1/2
}

// ---------------- WMMA GEMM: support = x @ W.T + b; h = support*d; msg = h --------

__global__ __launch_bounds__(256) void k_gemm_wmma(
    const float* __restrict__ x, const float* __restrict__ W,
    const float* __restrict__ bias, const float* __restrict__ d,
    float* __restrict__ support, float* __restrict__ h, float* __restrict__ msg)
{
    __shared__ float Wl[D * D];                // 16 KB of the 320 KB/WGP LDS
    for (int i = threadIdx.x; i < D * D; i += 256) Wl[i] = W[i];
    __syncthreads();

    const int wave = threadIdx.x >> 5;         // wave32: 8 waves / block
    const int lane = threadIdx.x & 31;
    const int tile = blockIdx.x * 8 + wave;    // one 16-row tile per wave
    if (tile >= N_TILES) return;               // wave-uniform: EXEC stays all-ones

    const int row0 = tile * 16;
    const int half = lane >> 4;                // 0: K lo pair, 1: K hi pair
    const int l15  = lane & 15;                // M (for A) / N (for B,C,D)

    v8f acc[4] = {v8f{}, v8f{}, v8f{}, v8f{}};
    const float* xrow = x + (size_t)(row0 + l15) * D;

    // K = 64 as 16 steps of V_WMMA_F32_16X16X4_F32
    #pragma unroll
    for (int kk = 0; kk < 16; ++kk) {
        const int k = kk * 4 + half * 2;
        // A 16x4 fp32 fragment: lane(0-15)=M holds K={k,k+1}; lanes 16-31 K={k+2,k+3}
        v2f a = *(const v2f*)(xrow + k);
        #pragma unroll
        for (int nt = 0; nt < 4; ++nt) {
            const int n = nt * 16 + l15;
            // B 4x16 fragment: B[k][n] = W[n][k] (GEMM is x @ W^T)
            v2f bf = *(const v2f*)(&Wl[n * D + k]);
            acc[nt] = __builtin_amdgcn_wmma_f32_16x16x4_f32(
                /*neg_a=*/false, a, /*neg_b=*/false, bf,
                /*c_mod=*/(short)0, acc[nt],
                /*reuse_a=*/false, /*reuse_b=*/false);
        }
    }

    // Epilogue: C/D layout — VGPR r, lanes 0-15: M=r; lanes 16-31: M=r+8; N=l15
    #pragma unroll
    for (int nt = 0; nt < 4; ++nt) {
        const int n  = nt * 16 + l15;
        const float bv = bias[n];
        #pragma unroll
        for (int r = 0; r < 8; ++r) {
            const int row  = row0 + r + 8 * half;
            const float s  = acc[nt][r] + bv;          // support
            const float hv = s * d[row];               // h = D^-1/2 * support
            const size_t idx = (size_t)row * D + n;
            support[idx] = s;
            h[idx]       = hv;
            msg[idx]     = hv;                         // identity term of (A+I)h
        }
    }
}

// ---------------- edge scatter: msg[row] += h[col] (L2-resident atomics) ----------

__global__ __launch_bounds__(256) void k_scatter(
    const int* __restrict__ erow, const int* __restrict__ ecol,
    const float* __restrict__ h, float* __restrict__ msg)
{
    const long long t = (long long)blockIdx.x * blockDim.x + threadIdx.x;
    const int e = (int)(t >> 6);       // edge
    const int j = (int)(t & 63);       // feature
    if (e < N_EDGES) {
        const int r = erow[e];
        const int c = ecol[e];
        atomicAdd(&msg[(size_t)r * D + j], h[(size_t)c * D + j]);
    }
}

// ---------------- finalize: out = (msg*d*0.5 + support)/1.5 ----------------------

__global__ void k_final(const float* __restrict__ msg, const float* __restrict__ support,
                        const float* __restrict__ d, float* __restrict__ out)
{
    const int i = blockIdx.x * blockDim.x + threadIdx.x;
    if (i < N_NODES * D) {
        const int row = i >> 6;
        out[i] = (msg[i] * d[row] * SMOOTHF + support[i]) * (1.0f / (1.0f + SMOOTHF));
    }
}

// ---------------- launcher --------------------------------------------------------

extern "C" void kernel_launch(void* const* d_in, const int* in_sizes, int n_in,
                              void* d_out, int out_size, void* d_ws, size_t ws_size,
                              hipStream_t stream) {
    const float* x    = (const float*)d_in[0];   // [50000, 64]
    const float* W    = (const float*)d_in[1];   // [64, 64]
    const float* bias = (const float*)d_in[2];   // [64]
    const int*   erow = (const int*)d_in[3];     // [800000] (harness passes int32)
    const int*   ecol = (const int*)d_in[4];     // [800000]
    float* out = (float*)d_out;

    float* ws      = (float*)d_ws;
    float* support = ws;                                   // 3.2M floats
    float* h       = ws + (size_t)N_NODES * D;             // 3.2M floats
    float* msg     = ws + 2 * (size_t)N_NODES * D;         // 3.2M floats
    float* deg     = ws + 3 * (size_t)N_NODES * D;         // 50000 floats (deg -> d)

    k_zero_deg<<<(N_NODES + 255) / 256, 256, 0, stream>>>(deg);
    k_degree  <<<(N_EDGES + 255) / 256, 256, 0, stream>>>(ecol, deg);
    k_rsqrt   <<<(N_NODES + 255) / 256, 256, 0, stream>>>(deg);
    k_gemm_wmma<<<(N_TILES + 7) / 8, 256, 0, stream>>>(x, W, bias, deg, support, h, msg);
    k_scatter <<<(N_EDGES * 64) / 256, 256, 0, stream>>>(erow, ecol, h, msg);
    k_final   <<<(N_NODES * D + 255) / 256, 256, 0, stream>>>(msg, support, deg, out);
}